// frequencyFeature_44702019616810
// MI455X (gfx1250) — compile-verified
//
#include <hip/hip_runtime.h>
#include <hip/hip_bf16.h>
#include <math.h>

typedef float v2f __attribute__((ext_vector_type(2)));
typedef float v8f __attribute__((ext_vector_type(8)));

#define NHW  512
#define KF   16      // padded low-frequency rank (true rank 11: i,j in 0..10)
#define LPAD 4       // LDS row padding (floats) to kill bank conflicts

__device__ __forceinline__ v8f wmma_f32_4(v2f a, v2f b, v8f c) {
  // D(16x16,f32) = A(16x4,f32) * B(4x16,f32) + C ; 8-arg VOP3P form
  return __builtin_amdgcn_wmma_f32_16x16x4_f32(false, a, false, b, (short)0, c,
                                               false, false);
}

// ---------------------------------------------------------------------------
// Kernel 1: build padded DCT / inverse-DCT matrices in workspace.
//   A[i][n] = (i<=10) ? 2*cos(pi*(2n+1)*i/(2N))        : 0   (16 x 512)
//   D[n][i] = (i<=10) ? w_i*cos(pi*(2n+1)*i/(2N))/N    : 0   (512 x 16), w_0=0.5
// ---------------------------------------------------------------------------
__global__ void init_consts(float* __restrict__ A, float* __restrict__ D) {
  int idx = blockIdx.x * blockDim.x + threadIdx.x;
  if (idx < KF * NHW) {
    int i = idx / NHW, n = idx % NHW;
    double arg = M_PI * (double)((2 * n + 1) * i) / (2.0 * NHW);
    A[idx] = (i <= 10) ? (float)(2.0 * cos(arg)) : 0.0f;
  }
  if (idx < NHW * KF) {
    int n = idx / KF, i = idx % KF;
    double w = (i == 0) ? 0.5 : 1.0;
    double arg = M_PI * (double)((2 * n + 1) * i) / (2.0 * NHW);
    D[idx] = (i <= 10) ? (float)(w * cos(arg) / (double)NHW) : 0.0f;
  }
}

// ---------------------------------------------------------------------------
// Kernel 2: per (image, 16-row tile): Y_part = A_tile * (X_tile * A^T)
// One wave32 per block. X tile staged in LDS (stride 516 -> conflict-free
// column reads for the A-fragment of the WMMA).
// ---------------------------------------------------------------------------
__global__ __launch_bounds__(32) void pass1_dct_low(const float* __restrict__ im,
                                                    const float* __restrict__ A,
                                                    float* __restrict__ Yp) {
  const int tile = blockIdx.x;                    // row tile [0,32)
  const int img  = blockIdx.y;
  const int lane = threadIdx.x;                   // 0..31, one wave
  const float* X = im + (size_t)img * NHW * NHW + (size_t)tile * 16 * NHW;

  __shared__ float Xs[16][NHW + LPAD];
  __shared__ float Ts[16][16 + LPAD];

  // coalesced b128 stage of the 16x512 tile into LDS
  #pragma unroll
  for (int r = 0; r < 16; ++r) {
    const float4* src = (const float4*)(X + r * NHW);
    float4* dst = (float4*)(&Xs[r][0]);           // row base 2064B -> 16B aligned
    for (int c = lane; c < NHW / 4; c += 32) dst[c] = src[c];
  }
  __syncthreads();

  const int m    = lane & 15;
  const int hi   = lane >> 4;                     // 0: lanes 0-15, 1: lanes 16-31
  const int kof  = hi * 2;                        // K offset of this half-wave
  const int row0 = hi * 8;                        // D-layout row base

  // T[h][j] = sum_w X[h][w] * A[j][w]   (16x16, K=512, 128 WMMAs)
  v8f t = {};
  for (int w0 = 0; w0 < NHW; w0 += 4) {
    v2f av, bv;
    av.x = Xs[m][w0 + kof];                       // A-frag: X[h=m][w0+kof..+1]
    av.y = Xs[m][w0 + kof + 1];
    bv.x = A[m * NHW + w0 + kof];                 // B-frag: B[k][n]=A[n][w0+k]
    bv.y = A[m * NHW + w0 + kof + 1];
    t = wmma_f32_4(av, bv, t);
  }

  // D-layout -> LDS so T can be re-read as a B fragment
  #pragma unroll
  for (int r = 0; r < 8; ++r) Ts[row0 + r][m] = t[r];
  __syncthreads();

  // Y_part[i][j] = sum_h A[i][tile*16+h] * T[h][j]   (K=16, 4 WMMAs)
  v8f y = {};
  #pragma unroll
  for (int k0 = 0; k0 < 16; k0 += 4) {
    const int kk = k0 + kof;
    v2f av, bv;
    av.x = A[m * NHW + tile * 16 + kk];
    av.y = A[m * NHW + tile * 16 + kk + 1];
    bv.x = Ts[kk][m];
    bv.y = Ts[kk + 1][m];
    y = wmma_f32_4(av, bv, y);
  }

  float* outp = Yp + ((size_t)img * 32 + tile) * 256;
  #pragma unroll
  for (int r = 0; r < 8; ++r) outp[(row0 + r) * 16 + m] = y[r];
}

// ---------------------------------------------------------------------------
// Kernel 3: deterministic reduction of the 32 per-tile partials per image.
// ---------------------------------------------------------------------------
__global__ void reduce_y(const float* __restrict__ Yp, float* __restrict__ Y) {
  const int img = blockIdx.x;
  const int e   = threadIdx.x;                    // 0..255
  float s = 0.0f;
  #pragma unroll 4
  for (int t = 0; t < 32; ++t) s += Yp[((size_t)img * 32 + t) * 256 + e];
  Y[(size_t)img * 256 + e] = s;
}

// ---------------------------------------------------------------------------
// Kernel 4: out = im - D_tile * (mask ⊙ Y) * D^T, per (image, 16-row tile).
// ---------------------------------------------------------------------------
__global__ __launch_bounds__(32) void pass2_reconstruct(const float* __restrict__ im,
                                                        const float* __restrict__ D,
                                                        const float* __restrict__ Y,
                                                        float* __restrict__ out) {
  const int tile = blockIdx.x;
  const int img  = blockIdx.y;
  const int lane = threadIdx.x;
  const int m    = lane & 15;
  const int hi   = lane >> 4;
  const int kof  = hi * 2;
  const int row0 = hi * 8;

  __shared__ float Ys[16][16 + LPAD];
  __shared__ float Ws[16][16 + LPAD];

  // masked low-frequency coefficients: KEEP i+j<=10 (that is what got removed
  // from the reference's spectrum, so out = im - reconstruct(kept))
  {
    const float* Yi = Y + (size_t)img * 256;
    #pragma unroll
    for (int r = 0; r < 8; ++r) {
      const int i = row0 + r;
      const float v = Yi[i * 16 + m];
      Ys[i][m] = (i + m <= 10) ? v : 0.0f;
    }
  }
  __syncthreads();

  // W[h][j] = sum_i D[tile*16+h][i] * Ym[i][j]   (K=16, 4 WMMAs)
  v8f w = {};
  #pragma unroll
  for (int k0 = 0; k0 < 16; k0 += 4) {
    const int kk = k0 + kof;
    v2f av, bv;
    av.x = D[(tile * 16 + m) * KF + kk];
    av.y = D[(tile * 16 + m) * KF + kk + 1];
    bv.x = Ys[kk][m];
    bv.y = Ys[kk + 1][m];
    w = wmma_f32_4(av, bv, w);
  }
  #pragma unroll
  for (int r = 0; r < 8; ++r) Ws[row0 + r][m] = w[r];
  __syncthreads();

  const size_t base = (size_t)img * NHW * NHW + (size_t)tile * 16 * NHW;
  const float* X = im + base;
  float* O = out + base;

  // Z[h][w] = sum_j W[h][j] * D[w][j] ; out = im - Z, 16 cols per step
  for (int w0 = 0; w0 < NHW; w0 += 16) {
    v8f z = {};
    #pragma unroll
    for (int k0 = 0; k0 < 16; k0 += 4) {
      const int kk = k0 + kof;
      v2f av, bv;
      av.x = Ws[m][kk];
      av.y = Ws[m][kk + 1];
      bv.x = D[(w0 + m) * KF + kk];
      bv.y = D[(w0 + m) * KF + kk + 1];
      z = wmma_f32_4(av, bv, z);
    }
    #pragma unroll
    for (int r = 0; r < 8; ++r) {
      const int idx = (row0 + r) * NHW + w0 + m;  // D-layout element (M,N)
      O[idx] = X[idx] - z[r];
    }
  }
}

// ---------------------------------------------------------------------------
extern "C" void kernel_launch(void* const* d_in, const int* in_sizes, int n_in,
                              void* d_out, int out_size, void* d_ws, size_t ws_size,
                              hipStream_t stream) {
  (void)n_in; (void)out_size; (void)ws_size;
  const float* im = (const float*)d_in[0];
  float* out = (float*)d_out;
  const int nimg = in_sizes[0] / (NHW * NHW);     // B*C = 96

  char*  ws  = (char*)d_ws;
  float* dA  = (float*)(ws);                                   // 16*512 f32 (32 KB)
  float* dD  = (float*)(ws + 32768);                           // 512*16 f32 (32 KB)
  float* dYp = (float*)(ws + 65536);                           // nimg*32*256 f32
  float* dY  = (float*)(ws + 65536 + (size_t)nimg * 32 * 256 * sizeof(float));

  init_consts<<<32, 256, 0, stream>>>(dA, dD);
  pass1_dct_low<<<dim3(32, nimg), 32, 0, stream>>>(im, dA, dYp);
  reduce_y<<<nimg, 256, 0, stream>>>(dYp, dY);
  pass2_reconstruct<<<dim3(32, nimg), 32, 0, stream>>>(im, dD, dY, out);
}